// SCAN_33586644254987
// MI455X (gfx1250) — compile-verified
//
#include <hip/hip_runtime.h>
#include <hip/hip_bf16.h>
#include <math.h>

typedef float v2f __attribute__((ext_vector_type(2)));
typedef float v8f __attribute__((ext_vector_type(8)));
typedef unsigned int v4u __attribute__((ext_vector_type(4)));
typedef int v4i __attribute__((ext_vector_type(4)));
typedef int v8i __attribute__((ext_vector_type(8)));

namespace {
constexpr int kR = 36, kT = 64, kD = 1024;
constexpr float kEps = 1e-8f;

// img in LDS: TDM pads 1 DWORD every 256 -> row stride 1028 (1028 % 64 == 4, conflict-free
// column reads). Element (m,k) lives at m*IMG_LD + k + (k>>8).
constexpr int IMG_LD = 1028;
constexpr int CAP_CK = 128;    // cap K-chunk width
// cap chunk in LDS: TDM pads 4 DWORDs every 128 -> row stride 132 (132 % 64 == 4).
constexpr int CAP_LD = 132;
constexpr int ATT_LD = 65;     // attn row stride

// LDS layout in floats
constexpr int SM_IMG   = 0;
constexpr int SM_CAP0  = SM_IMG  + kR * IMG_LD;
constexpr int SM_CAP1  = SM_CAP0 + kT * CAP_LD;
constexpr int SM_SC    = SM_CAP1 + kT * CAP_LD;     // 48 x 64 dots
constexpr int SM_ATT   = SM_SC   + 48 * 64;         // 36 x ATT_LD attn
constexpr int SM_INP   = SM_ATT  + kR * ATT_LD;     // img-norm partials 36x4
constexpr int SM_CNP   = SM_INP  + kR * 4;          // cap-norm partials 64x4
constexpr int SM_INORM = SM_CNP  + kT * 4;          // img norms (36)
constexpr int SM_TOTAL = SM_INORM + kR;
constexpr size_t kSmemBytes = (size_t)SM_TOTAL * sizeof(float);
} // namespace

// k -> padded column offset inside an img LDS row
__device__ __forceinline__ int IC(int k) { return k + (k >> 8); }

// Issue a TDM 2D tile load Global->LDS. Dimensions in 4-byte elements.
// pad_interval/pad_amount use the D# encodings (interval: 0=2..7=256 DW; amount: n -> n+1 DW).
__device__ __forceinline__ void tdm_load_2d(const void* gsrc, unsigned lds_byte_addr,
                                            unsigned tensor_d0, unsigned tensor_d1,
                                            unsigned stride_d0,
                                            unsigned tile_d0, unsigned tile_d1,
                                            bool pad_en, unsigned pad_interval,
                                            unsigned pad_amount) {
  unsigned long long ga = (unsigned long long)gsrc;
  v4u g0;
  g0[0] = 1u;                                          // count=1, user descriptor
  g0[1] = lds_byte_addr;                               // lds_addr
  g0[2] = (unsigned)(ga & 0xFFFFFFFFu);                // global_addr[31:0]
  g0[3] = (unsigned)((ga >> 32) & 0x1FFFFFFu) | (2u << 30);  // addr[56:32] | type=2
  v8i g1;
  unsigned w0 = (2u << 16);                            // data_size = 4B; wg_mask = 0
  if (pad_en) w0 |= (1u << 20) | (pad_interval << 22) | (pad_amount << 25);
  g1[0] = (int)w0;
  g1[1] = (int)((tensor_d0 & 0xFFFFu) << 16);                      // dim0[15:0] @ bits63:48
  g1[2] = (int)(((tensor_d0 >> 16) & 0xFFFFu) | ((tensor_d1 & 0xFFFFu) << 16));
  g1[3] = (int)(((tensor_d1 >> 16) & 0xFFFFu) | (tile_d0 << 16));  // dim1 hi | tile_dim0
  g1[4] = (int)(tile_d1 & 0xFFFFu);                                // tile_dim1; tile_dim2=0
  g1[5] = (int)stride_d0;                                          // dim0_stride[31:0]
  g1[6] = 0;                                                       // stride hi | dim1_stride lo
  g1[7] = 0;
  v4i g2 = {0, 0, 0, 0};
  v4i g3 = {0, 0, 0, 0};
#if defined(__clang_major__) && (__clang_major__ >= 23)
  v8i g4 = {0, 0, 0, 0, 0, 0, 0, 0};
  __builtin_amdgcn_tensor_load_to_lds(g0, g1, g2, g3, g4, 0);
#else
  __builtin_amdgcn_tensor_load_to_lds(g0, g1, g2, g3, 0);
#endif
}

__device__ __forceinline__ unsigned lds_off(const void* p) {
  return (unsigned)(unsigned long long)p;   // low 32 bits of generic shared ptr = LDS addr
}

__device__ __forceinline__ v8f wmma_f32(v2f a, v2f b, v8f c) {
  return __builtin_amdgcn_wmma_f32_16x16x4_f32(false, a, false, b, (short)0, c, false, false);
}

__global__ __launch_bounds__(256, 1)
void SCAN_33586644254987_kernel(const float* __restrict__ img,
                                const float* __restrict__ cap,
                                float* __restrict__ out) {
  extern __shared__ float smem[];
  float* simg   = smem + SM_IMG;
  float* scap0  = smem + SM_CAP0;
  float* scap1  = smem + SM_CAP1;
  float* ssc    = smem + SM_SC;
  float* satt   = smem + SM_ATT;
  float* sinp   = smem + SM_INP;
  float* scnp   = smem + SM_CNP;
  float* sinorm = smem + SM_INORM;

  const int b     = blockIdx.x;
  const int tid   = threadIdx.x;
  const int wid   = tid >> 5;        // wave id 0..7 (wave32)
  const int lane  = tid & 31;
  const int lhalf = lane >> 4;
  const int l16   = lane & 15;

  const float* gimg = img + (size_t)b * kR * kD;
  const float* gcap = cap + (size_t)b * kT * kD;
  float*       gout = out + (size_t)b * kT * kD;

  // ---- TDM: start img tile (36x1024, pad->stride 1028) + first cap chunk ----
  if (wid == 0) {
    tdm_load_2d(gimg, lds_off(simg), kD, kR, kD, kD, kR,
                /*pad*/true, /*interval 256DW*/7, /*amount 1DW*/0);
    tdm_load_2d(gcap, lds_off(scap0), kD, kT, kD, CAP_CK, kT,
                /*pad*/true, /*interval 128DW*/6, /*amount 4DW*/3);
    __builtin_amdgcn_s_wait_tensorcnt(1);   // img complete (in-order per wave)
  }
  scnp[tid] = 0.0f;                         // cap-norm partials (64x4)
  __syncthreads();

  // ---- img row norms: 4 threads per row (data contiguous at part*257) ----
  if (tid < kR * 4) {
    int row = tid >> 2, part = tid & 3;
    const float* p = &simg[row * IMG_LD + part * 257];
    float s = 0.0f;
    for (int c = 0; c < 256; ++c) { float x = p[c]; s += x * x; }
    sinp[tid] = s;
  }
  __syncthreads();
  if (tid < kR) {
    sinorm[tid] = sqrtf(sinp[tid * 4] + sinp[tid * 4 + 1] +
                        sinp[tid * 4 + 2] + sinp[tid * 4 + 3]);
  }

  // ---- phase 1: dots(48x64) = img x cap^T; cap chunks TDM double-buffered ----
  // Wave w owns tile w (rows mi0*16.., always valid) and, for w<4, tile w+8
  // (rows 32..47, same N-columns as tile w -> shared B fragment).
  const int mi0    = wid >> 2;
  const int a0base = (mi0 * 16 + l16) * IMG_LD;
  // tile1 A rows: lanes l16>=4 map to padded rows 36..47; clamp their address to the
  // (valid, in-bounds) row 32 and zero the value with a multiplicative mask instead of
  // a conditional load (avoids per-load exec save/restore).
  const int  a1base = ((l16 < 4) ? (32 + l16) : 32) * IMG_LD;
  const float m1    = (l16 < 4) ? 1.0f : 0.0f;
  const int nrow    = ((wid & 3) * 16 + l16) * CAP_LD;
  v8f acc0 = {}, acc1 = {};

  for (int c = 0; c < kD / CAP_CK; ++c) {
    if (wid == 0) __builtin_amdgcn_s_wait_tensorcnt(0);   // chunk c arrived
    __syncthreads();                                      // visible to all; buf[c-1] free
    if (wid == 0 && c + 1 < kD / CAP_CK) {
      tdm_load_2d(gcap + (c + 1) * CAP_CK, lds_off((c & 1) ? scap0 : scap1),
                  kD, kT, kD, CAP_CK, kT, true, 6, 3);     // overlaps compute below
    }
    const float* sbuf = (c & 1) ? scap1 : scap0;
    const int kc = c * CAP_CK;

    // cap-norm partial accumulation (all 256 threads; 64 rows x 4 parts x 32 cols)
    {
      int row = tid >> 2, part = tid & 3;
      const float* p = &sbuf[row * CAP_LD + part * 32];
      float s = 0.0f;
      for (int cc = 0; cc < 32; ++cc) { float x = p[cc]; s += x * x; }
      scnp[tid] += s;
    }

    if (wid < 4) {
#pragma unroll 4
      for (int kk = 0; kk < CAP_CK; kk += 4) {
        int ks = kk + 2 * lhalf;
        int kg = kc + ks;
        v2f bb; bb.x = sbuf[nrow + ks]; bb.y = sbuf[nrow + ks + 1];
        v2f a0; a0.x = simg[a0base + IC(kg)]; a0.y = simg[a0base + IC(kg + 1)];
        v2f a1;
        a1.x = simg[a1base + IC(kg)]     * m1;   // unconditional load + mask
        a1.y = simg[a1base + IC(kg + 1)] * m1;
        acc0 = wmma_f32(a0, bb, acc0);
        acc1 = wmma_f32(a1, bb, acc1);
      }
    } else {
#pragma unroll 4
      for (int kk = 0; kk < CAP_CK; kk += 4) {
        int ks = kk + 2 * lhalf;
        int kg = kc + ks;
        v2f bb; bb.x = sbuf[nrow + ks]; bb.y = sbuf[nrow + ks + 1];
        v2f a0; a0.x = simg[a0base + IC(kg)]; a0.y = simg[a0base + IC(kg + 1)];
        acc0 = wmma_f32(a0, bb, acc0);
      }
    }
  }

  // write dots tiles to LDS (D layout: VGPR j -> M = j + 8*lhalf)
#pragma unroll
  for (int j = 0; j < 8; ++j) {
    int row = mi0 * 16 + j + 8 * lhalf;
    ssc[row * 64 + (wid & 3) * 16 + l16] = acc0[j];
  }
  if (wid < 4) {
#pragma unroll
    for (int j = 0; j < 8; ++j) {
      int row = 32 + j + 8 * lhalf;
      ssc[row * 64 + wid * 16 + l16] = acc1[j];
    }
  }
  __syncthreads();   // dots + cap-norm partials complete

  // ---- phase 2: cosine scores + softmax over regions (one thread per word t) ----
  if (tid < kT) {
    int t = tid;
    float cn = sqrtf(scnp[t * 4] + scnp[t * 4 + 1] + scnp[t * 4 + 2] + scnp[t * 4 + 3]);
    float sc[kR];
    float mx = -3.4e38f;
#pragma unroll
    for (int r = 0; r < kR; ++r) {
      float s = ssc[r * 64 + t] / fmaxf(sinorm[r] * cn, kEps);
      sc[r] = s;
      mx = fmaxf(mx, s);
    }
    float sum = 0.0f;
#pragma unroll
    for (int r = 0; r < kR; ++r) { float e = expf(sc[r] - mx); sc[r] = e; sum += e; }
    float inv = 1.0f / sum;
#pragma unroll
    for (int r = 0; r < kR; ++r) satt[r * ATT_LD + t] = sc[r] * inv;
  }
  __syncthreads();

  // ---- phase 3: out(64x1024) = attn^T x img ----
  // A fragments (attn, K=36) hoisted to registers once; per column group the
  // single B fragment feeds 4 independent accumulators (mi = 0..3).
  v2f Af[9][4];
#pragma unroll
  for (int k9 = 0; k9 < 9; ++k9) {
    int kr = 4 * k9 + 2 * lhalf;
#pragma unroll
    for (int mi = 0; mi < 4; ++mi) {
      Af[k9][mi].x = satt[kr * ATT_LD + mi * 16 + l16];
      Af[k9][mi].y = satt[(kr + 1) * ATT_LD + mi * 16 + l16];
    }
  }
  for (int nb = wid * 16; nb < kD; nb += 128) {
    v8f o0 = {}, o1 = {}, o2 = {}, o3 = {};
    const int col = IC(nb + l16);   // never crosses a pad boundary within a 16-group
#pragma unroll
    for (int k9 = 0; k9 < 9; ++k9) {
      int kr = 4 * k9 + 2 * lhalf;
      v2f bb;
      bb.x = simg[kr * IMG_LD + col];
      bb.y = simg[(kr + 1) * IMG_LD + col];
      o0 = wmma_f32(Af[k9][0], bb, o0);
      o1 = wmma_f32(Af[k9][1], bb, o1);
      o2 = wmma_f32(Af[k9][2], bb, o2);
      o3 = wmma_f32(Af[k9][3], bb, o3);
    }
    int d = nb + l16;
#pragma unroll
    for (int j = 0; j < 8; ++j) {
      int r = j + 8 * lhalf;
      gout[r * kD + d]        = o0[j];
      gout[(16 + r) * kD + d] = o1[j];
      gout[(32 + r) * kD + d] = o2[j];
      gout[(48 + r) * kD + d] = o3[j];
    }
  }
}

extern "C" void kernel_launch(void* const* d_in, const int* in_sizes, int n_in,
                              void* d_out, int out_size, void* d_ws, size_t ws_size,
                              hipStream_t stream) {
  (void)in_sizes; (void)n_in; (void)d_ws; (void)ws_size; (void)out_size;
  const float* img = (const float*)d_in[0];   // (256, 36, 1024) f32
  const float* cap = (const float*)d_in[1];   // (256, 64, 1024) f32
  float* out = (float*)d_out;                 // (256, 64, 1024) f32

  // ~239 KB dynamic LDS per workgroup (WGP has 320 KB).
  (void)hipFuncSetAttribute((const void*)SCAN_33586644254987_kernel,
                            hipFuncAttributeMaxDynamicSharedMemorySize,
                            (int)kSmemBytes);
  SCAN_33586644254987_kernel<<<256, 256, kSmemBytes, stream>>>(img, cap, out);
}